// ARMAConv_936302871076
// MI455X (gfx1250) — compile-verified
//
#include <hip/hip_runtime.h>
#include <cstdint>
#include <cstddef>

// ---------------------------------------------------------------------------
// ARMAConv (2 layers, 2 identical stacks) on MI455X (gfx1250).
//
// CSR built on-device (counting sort + LDS prefix scan), then two atomic-free
// gather passes. Edge (col,norm) pairs are interleaved and streamed into LDS
// with the CDNA5 async path (global_load_async_to_lds_b64 + s_wait_asynccnt),
// double-buffered so the next 32-edge chunk is in flight while the current one
// is consumed. One wave32 per node; each lane owns a float2 channel pair.
//
// WMMA intentionally unused: sparse SpMM at ~0.1 FLOP/byte is L2-bandwidth
// bound; dense-tile reformulation would inflate FLOPs ~16x for no gain.
// ---------------------------------------------------------------------------

static constexpr int   CH     = 64;   // channels
static constexpr int   SCAN_B = 1024; // scan block size

// Async global->LDS issue: drain DS first (async LDS writes are unordered vs
// DS reads of a previously consumed chunk -> WAR hazard on the staging half).
#define ASYNC_PAIR_LOAD(ldsaddr, gaddr)                                   \
  asm volatile("s_wait_dscnt 0\n\t"                                       \
               "global_load_async_to_lds_b64 %0, %1, off"                 \
               :: "v"(ldsaddr), "v"(gaddr) : "memory")

// ---- K0: zero a u32 array ----
__global__ void k_zero_u32(uint32_t* __restrict__ p, int n) {
  int i = blockIdx.x * blockDim.x + threadIdx.x;
  if (i < n) p[i] = 0u;
}

// ---- K1: degree count (deg[row[e]]++) ----
__global__ void k_degree(const int* __restrict__ row, int E,
                         uint32_t* __restrict__ deg) {
  int i = blockIdx.x * blockDim.x + threadIdx.x;
  if (i < E) atomicAdd(&deg[row[i]], 1u);
}

// ---- K2: dis = deg>0 ? rsqrt(deg) : 0 ----
__global__ void k_dis(const uint32_t* __restrict__ deg,
                      float* __restrict__ dis, int N) {
  int i = blockIdx.x * blockDim.x + threadIdx.x;
  if (i < N) {
    uint32_t d = deg[i];
    dis[i] = d ? rsqrtf((float)d) : 0.0f;
  }
}

// ---- K3: per-block sums of deg (LDS tree reduce) ----
__global__ void __launch_bounds__(SCAN_B)
k_blocksum(const uint32_t* __restrict__ deg, int N,
           uint32_t* __restrict__ partials) {
  __shared__ uint32_t sm[SCAN_B];
  int t = threadIdx.x;
  int i = blockIdx.x * SCAN_B + t;
  sm[t] = (i < N) ? deg[i] : 0u;
  __syncthreads();
  for (int s = SCAN_B / 2; s > 0; s >>= 1) {
    if (t < s) sm[t] += sm[t + s];
    __syncthreads();
  }
  if (t == 0) partials[blockIdx.x] = sm[0];
}

// ---- K4: single-block exclusive scan of partials (P <= SCAN_B) ----
__global__ void __launch_bounds__(SCAN_B)
k_scan_partials(uint32_t* __restrict__ partials, int P) {
  __shared__ uint32_t sm[SCAN_B];
  int t = threadIdx.x;
  uint32_t v = (t < P) ? partials[t] : 0u;
  sm[t] = v;
  __syncthreads();
  for (int off = 1; off < SCAN_B; off <<= 1) {
    uint32_t u = (t >= off) ? sm[t - off] : 0u;
    __syncthreads();
    sm[t] += u;
    __syncthreads();
  }
  if (t < P) partials[t] = sm[t] - v;  // exclusive
}

// ---- K5: final exclusive scan -> CSR row offsets ----
__global__ void __launch_bounds__(SCAN_B)
k_scan_final(const uint32_t* __restrict__ deg, int N,
             const uint32_t* __restrict__ partials,
             uint32_t* __restrict__ offs) {
  __shared__ uint32_t sm[SCAN_B];
  int t = threadIdx.x;
  int i = blockIdx.x * SCAN_B + t;
  uint32_t v = (i < N) ? deg[i] : 0u;
  sm[t] = v;
  __syncthreads();
  for (int off = 1; off < SCAN_B; off <<= 1) {
    uint32_t u = (t >= off) ? sm[t - off] : 0u;
    __syncthreads();
    sm[t] += u;
    __syncthreads();
  }
  uint32_t ex = partials[blockIdx.x] + sm[t] - v;
  if (i < N) offs[i] = ex;
  if (i == N - 1) offs[N] = ex + v;  // == E
}

// ---- K6: scatter edges into CSR as interleaved (col, norm) pairs ----
__global__ void k_fill(const int* __restrict__ row, const int* __restrict__ col,
                       int E, const float* __restrict__ dis,
                       const uint32_t* __restrict__ offs,
                       uint32_t* __restrict__ cursor,
                       uint32_t* __restrict__ pair) {
  int i = blockIdx.x * blockDim.x + threadIdx.x;
  if (i >= E) return;
  int r = row[i];
  int c = col[i];
  uint32_t pos = offs[r] + atomicAdd(&cursor[r], 1u);
  pair[2u * pos]      = (uint32_t)c;
  pair[2u * pos + 1u] = __float_as_uint(dis[r] * dis[c]);
}

// ---- K7: propagate out[i,:] = sum_e norm_e * xin[col_e,:]  (wave per node)
// Double-buffered async edge staging: chunk t+1 streams into the other LDS
// half while chunk t is consumed. s_wait_asynccnt 1 suffices because async
// loads complete in order within a wave.
__global__ void __launch_bounds__(256)
k_prop(const float* __restrict__ xin, int N,
       const uint32_t* __restrict__ offs,
       const uint32_t* __restrict__ pair,
       float* __restrict__ out) {
  __shared__ __align__(16) uint32_t stage[8 * 128];  // 8 waves x 2 halves x 32 pairs
  const int lane = threadIdx.x & 31;
  const int warp = threadIdx.x >> 5;
  const int node = blockIdx.x * 8 + warp;
  if (node >= N) return;                     // wave-uniform exit

  const uint32_t s = offs[node];
  const uint32_t e = offs[node + 1];
  const float2* __restrict__ xv = (const float2*)xin;
  uint32_t* half0 = &stage[warp * 128];
  uint32_t* half1 = half0 + 64;
  const uint32_t lds0 = (uint32_t)(uintptr_t)(&half0[lane * 2]);
  const uint32_t lds1 = (uint32_t)(uintptr_t)(&half1[lane * 2]);

  const uint32_t nchunk = (e - s + 31u) >> 5;   // 0 for isolated nodes
  if (nchunk) {                                 // prime the pipeline
    uint32_t i0 = s + (uint32_t)lane;
    if (i0 >= e) i0 = e - 1u;
    ASYNC_PAIR_LOAD(lds0, (uint64_t)(uintptr_t)(pair + 2u * i0));
  }

  float ax = 0.0f, ay = 0.0f;
  for (uint32_t t = 0; t < nchunk; ++t) {
    const uint32_t* buf = (t & 1u) ? half1 : half0;
    if (t + 1u < nchunk) {
      uint32_t in = s + (t + 1u) * 32u + (uint32_t)lane;
      if (in >= e) in = e - 1u;
      ASYNC_PAIR_LOAD(((t + 1u) & 1u) ? lds1 : lds0,
                      (uint64_t)(uintptr_t)(pair + 2u * in));
      asm volatile("s_wait_asynccnt 1" ::: "memory");  // oldest (chunk t) done
    } else {
      asm volatile("s_wait_asynccnt 0" ::: "memory");
    }
    const uint32_t base = s + t * 32u;
    const int cnt = (int)min(32u, e - base);
    for (int j = 0; j < cnt; ++j) {
      uint32_t c = buf[2 * j];                       // broadcast LDS read
      float    w = __uint_as_float(buf[2 * j + 1]);
      float2   v = xv[(size_t)c * 32 + lane];        // coalesced B64 gather
      ax = fmaf(w, v.x, ax);
      ay = fmaf(w, v.y, ay);
    }
  }
  float2* ov = (float2*)out;
  ov[(size_t)node * 32 + lane] = make_float2(ax, ay);
}

// ---- K8: cur = 0.9*acc + 0.1*x ----
__global__ void k_mix(const float* __restrict__ acc, const float* __restrict__ x,
                      float* __restrict__ dst, int n) {
  int i = blockIdx.x * blockDim.x + threadIdx.x;
  if (i < n) dst[i] = fmaf(0.9f, acc[i], 0.1f * x[i]);
}

// ---- K9: out = sum_s softmax(sw)_s * (0.9*out + 0.1*x)  (in-place) ----
__global__ void k_final(float* __restrict__ out, const float* __restrict__ x,
                        const float* __restrict__ sw, int S, int n) {
  int i = blockIdx.x * blockDim.x + threadIdx.x;
  if (i >= n) return;
  float m = sw[0];
  for (int s = 1; s < S; ++s) m = fmaxf(m, sw[s]);
  float denom = 0.0f;
  for (int s = 0; s < S; ++s) denom += __expf(sw[s] - m);
  float t = fmaf(0.9f, out[i], 0.1f * x[i]);
  float r = 0.0f;
  for (int s = 0; s < S; ++s) r += (__expf(sw[s] - m) / denom) * t;
  out[i] = r;
}

// ---------------------------------------------------------------------------
extern "C" void kernel_launch(void* const* d_in, const int* in_sizes, int n_in,
                              void* d_out, int out_size, void* d_ws, size_t ws_size,
                              hipStream_t stream) {
  const float* x  = (const float*)d_in[0];          // [N,64] f32
  const int*   ei = (const int*)d_in[1];            // [2,E] indices
  const float* sw = (const float*)d_in[2];          // [S]   f32
  const int N = in_sizes[0] / CH;
  const int E = in_sizes[1] / 2;
  const int S = in_sizes[2];
  const int* row = ei;
  const int* col = ei + E;
  float* out = (float*)d_out;

  // ---- workspace carve (4-byte words; keep 8B alignment for pair) ----
  uint32_t* p = (uint32_t*)d_ws;
  uint32_t* deg      = p;              p += N;          // N even
  float*    dis      = (float*)p;      p += N;
  uint32_t* offs     = p;              p += (N + 2);    // keep even
  uint32_t* cursor   = p;              p += N;
  uint32_t* partials = p;              p += SCAN_B;     // >= #scan blocks
  uint32_t* pair     = p;              p += 2 * (size_t)E;  // 8B-aligned
  float*    cur      = (float*)p;                       // N*64 f32

  const int nElem = N * CH;
  const int B = 256;
  const int gE  = (E + B - 1) / B;
  const int gN  = (N + B - 1) / B;
  const int gC  = (nElem + B - 1) / B;
  const int P   = (N + SCAN_B - 1) / SCAN_B;            // scan blocks (<=SCAN_B)
  const int gNd = (N + 7) / 8;                          // 8 waves per block

  // 1-2. degree
  k_zero_u32<<<gN, B, 0, stream>>>(deg, N);
  k_zero_u32<<<gN, B, 0, stream>>>(cursor, N);
  k_degree<<<gE, B, 0, stream>>>(row, E, deg);
  // 3. dis
  k_dis<<<gN, B, 0, stream>>>(deg, dis, N);
  // 4-6. exclusive prefix scan of deg -> offs
  k_blocksum<<<P, SCAN_B, 0, stream>>>(deg, N, partials);
  k_scan_partials<<<1, SCAN_B, 0, stream>>>(partials, P);
  k_scan_final<<<P, SCAN_B, 0, stream>>>(deg, N, partials, offs);
  // 7. CSR fill (interleaved col/norm pairs)
  k_fill<<<gE, B, 0, stream>>>(row, col, E, dis, offs, cursor, pair);
  // 8. layer 1: out = P(x);  cur = 0.9*out + 0.1*x
  k_prop<<<gNd, B, 0, stream>>>(x, N, offs, pair, out);
  k_mix<<<gC, B, 0, stream>>>(out, x, cur, nElem);
  // 9. layer 2: out = P(cur); out = softmax-sum * (0.9*out + 0.1*x)
  k_prop<<<gNd, B, 0, stream>>>(cur, N, offs, pair, out);
  k_final<<<gC, B, 0, stream>>>(out, x, sw, S, nElem);
}